// GibbsSampler_90443421319469
// MI455X (gfx1250) — compile-verified
//
#include <hip/hip_runtime.h>
#include <hip/hip_bf16.h>
#include <stdint.h>

#define LXD 256
#define LYD 256
#define NSWEEPS 2
#define NSITES (LXD * LYD)
#define CHUNK 32
#define NCHUNKS (NSITES / CHUNK)
#define NCHUNKS_TOT (NSWEEPS * NCHUNKS)
#define BLOCKT 256
#define NSTAGE 4  // async staging pipeline depth (chunks in flight)
#define MASK_BYTES ((size_t)NSITES * sizeof(unsigned))  // 256 KB: one mask/site

// ------------------------- capability gates (device pass only) ----------
#if defined(__AMDGCN__)
#if __has_builtin(__builtin_amdgcn_global_load_async_to_lds_b32)
#define HAVE_ASYNC32 1
#else
#define HAVE_ASYNC32 0
#endif
#if __has_builtin(__builtin_amdgcn_s_wait_asynccnt)
#define HAVE_WAITASYNC 1
#else
#define HAVE_WAITASYNC 0
#endif
#if __has_builtin(__builtin_amdgcn_tensor_load_to_lds)
#define HAVE_TDM 1
#else
#define HAVE_TDM 0
#endif
#if __has_builtin(__builtin_amdgcn_s_wait_tensorcnt)
#define HAVE_WAITTENSOR 1
#else
#define HAVE_WAITTENSOR 0
#endif
#else
#define HAVE_ASYNC32 0
#define HAVE_WAITASYNC 0
#define HAVE_TDM 0
#define HAVE_WAITTENSOR 0
#endif

// clang prints AS1 as "__device__", AS3 as "__shared__": the async builtins
// take typed int* in those address spaces (probe-verified round 1).
typedef __attribute__((address_space(1))) int as1_int;
typedef __attribute__((address_space(3))) int as3_int;
typedef unsigned int u32x4 __attribute__((ext_vector_type(4)));
typedef int i32x4 __attribute__((ext_vector_type(4)));
typedef int i32x8 __attribute__((ext_vector_type(8)));

__device__ __forceinline__ unsigned lds_offset_of(void* p) {
  // generic "shared" pointers carry the LDS byte offset in the low 32 bits
  return (unsigned)(unsigned long long)p;
}

__device__ __forceinline__ as1_int* to_as1(const void* p) {
  return (as1_int*)(unsigned long long)p;
}

__device__ __forceinline__ as3_int* to_as3(void* p) {
  return (as3_int*)lds_offset_of(p);
}

__device__ __forceinline__ void async_cp_b32(const void* gsrc, void* ldst) {
#if HAVE_ASYNC32
  __builtin_amdgcn_global_load_async_to_lds_b32(to_as1(gsrc), to_as3(ldst), 0, 0);
#else
  (void)gsrc; (void)ldst;
#endif
}

__device__ __forceinline__ void mem_fence_compiler() {
#if defined(__AMDGCN__)
  __builtin_amdgcn_wave_barrier();
#endif
  asm volatile("" ::: "memory");
}

// Tensor Data Mover: DMA one 32-element row of int32 (128B) from global into LDS.
__device__ __forceinline__ void tdm_load_row32_i32(const int* gsrc, void* ldst) {
#if HAVE_TDM && HAVE_WAITTENSOR
  unsigned long long ga = (unsigned long long)gsrc;
  u32x4 g0;
  g0[0] = 1u;                                   // count=1, user mode, no gather
  g0[1] = lds_offset_of(ldst);                  // lds_addr
  g0[2] = (unsigned)(ga & 0xFFFFFFFFu);         // global_addr[31:0]
  g0[3] = (unsigned)((ga >> 32) & 0x01FFFFFFu)  // global_addr[56:32]
          | (2u << 30);                         // type = 2 ("image")
  i32x8 g1;
  g1[0] = (int)(2u << 16);   // workgroup_mask=0 (not in cluster), data_size=2 (4B)
  g1[1] = (int)(32u << 16);  // tensor_dim0 = 32  (bits 79:48, low half)
  g1[2] = (int)(1u << 16);   // tensor_dim0 hi = 0, tensor_dim1 = 1 (low half)
  g1[3] = (int)(32u << 16);  // tensor_dim1 hi = 0, tile_dim0 = 32
  g1[4] = 0;                 // tile_dim1 = 0 (unused), tile_dim2 = 0
  g1[5] = 32;                // tensor_dim0_stride = 32
  g1[6] = 0;
  g1[7] = 0;
  i32x4 gz = {0, 0, 0, 0};
#if __clang_major__ >= 23
  i32x8 gz8 = {0, 0, 0, 0, 0, 0, 0, 0};
  __builtin_amdgcn_tensor_load_to_lds(g0, g1, gz, gz, gz8, 0);
#else
  __builtin_amdgcn_tensor_load_to_lds(g0, g1, gz, gz, 0);
#endif
  __builtin_amdgcn_s_wait_tensorcnt(0);
#else
  (void)gsrc; (void)ldst;
#endif
}

// ---------------------------------------------------------------------------
// Kernel A (parallel): per-site mask of earlier in-chunk conflicting lanes.
// Conflicts depend only on `perm`, so this is hoisted off the serial chain.
// ---------------------------------------------------------------------------
__global__ void __launch_bounds__(BLOCKT)
conflict_mask_kernel(const int* __restrict__ perm,
                     unsigned* __restrict__ masks) {
  const int lane = threadIdx.x & (CHUNK - 1);
  const int chunk = blockIdx.x * (BLOCKT / CHUNK) + (threadIdx.x >> 5);
  if (chunk >= NCHUNKS) return;
  const int i = chunk * CHUNK + lane;
  const int uv = perm[i];
  const int u = uv >> 8, v = uv & (LYD - 1);
  unsigned earlier = 0;
#pragma unroll 1
  for (int j = 0; j < CHUNK; ++j) {
    int uvj = __shfl(uv, j, CHUNK);
    int uj = uvj >> 8, vj = uvj & (LYD - 1);
    int du = u - uj; du = du < 0 ? -du : du;
    int dv = v - vj; dv = dv < 0 ? -dv : dv;
    if (j < lane && du <= 1 && dv <= 1) earlier |= (1u << j);
  }
  masks[i] = earlier;
}

// ---------------------------------------------------------------------------
// Kernel B (one wave32): the sequential Gibbs chain, lattice resident in LDS.
// ---------------------------------------------------------------------------
__global__ void __launch_bounds__(BLOCKT)
gibbs_potts_kernel(const int* __restrict__ X_init,
                   const int* __restrict__ perm,
                   const float* __restrict__ uniforms,
                   const float* __restrict__ beta_p,
                   const unsigned* __restrict__ masks,  // may be unused
                   const int use_masks,
                   float* __restrict__ out) {
  extern __shared__ unsigned char smem[];
  unsigned char* Xs = smem;  // 64 KB lattice, 1 B/site
  int* stP = (int*)(smem + NSITES);                                   // [NSTAGE][CHUNK]
  float* stU = (float*)(smem + NSITES + NSTAGE * CHUNK * 4);          // [NSTAGE][CHUNK]
  unsigned* stM = (unsigned*)(smem + NSITES + 2 * NSTAGE * CHUNK * 4);// [NSTAGE][CHUNK]

  const int tid = threadIdx.x;

  // Cooperative pack: int32 lattice -> byte lattice in LDS.
  for (int base = tid * 4; base < NSITES; base += BLOCKT * 4) {
    int4 w = *(const int4*)(X_init + base);
    unsigned packed = (unsigned)(w.x & 3) | ((unsigned)(w.y & 3) << 8) |
                      ((unsigned)(w.z & 3) << 16) | ((unsigned)(w.w & 3) << 24);
    *(unsigned*)(Xs + base) = packed;
  }
  __syncthreads();

  const float beta = beta_p[0];

  if (tid < CHUNK) {  // one wave32 owns the sequential Gibbs chain
    const int lane = tid;
#if defined(__AMDGCN__)
    __builtin_amdgcn_s_setprio(3);
#endif

    // TDM warms stage buffer 0 with chunk 0's permutation entries.
    tdm_load_row32_i32(perm, stP);

#if HAVE_ASYNC32 && HAVE_WAITASYNC
    // Prologue: fill NSTAGE-1 = 3 chunks of the staging pipeline.
#pragma unroll
    for (int k = 0; k < NSTAGE - 1; ++k) {
      async_cp_b32(perm + k * CHUNK + lane, stP + k * CHUNK + lane);
      async_cp_b32(uniforms + (size_t)k * CHUNK + lane, stU + k * CHUNK + lane);
      if (use_masks)
        async_cp_b32(masks + k * CHUNK + lane, stM + k * CHUNK + lane);
    }
#endif

    for (int t = 0; t < NCHUNKS_TOT; ++t) {
      const int c = t & (NCHUNKS - 1);  // chunk index within a sweep
      const int buf = t & (NSTAGE - 1);

      int uv;
      float r;
      unsigned earlier = 0;
      bool have_mask = false;
#if HAVE_ASYNC32 && HAVE_WAITASYNC
      // Async loads complete in order: waiting for <= 2 chunks' worth of
      // outstanding loads guarantees the oldest chunk (t) has landed while
      // keeping ~2 chunks of HBM latency hidden behind compute.
      if (use_masks) {
        __builtin_amdgcn_s_wait_asynccnt(6);  // 3 streams x 2 chunks in flight
      } else {
        __builtin_amdgcn_s_wait_asynccnt(4);  // 2 streams x 2 chunks in flight
      }
      mem_fence_compiler();
      uv = stP[buf * CHUNK + lane];
      r = stU[buf * CHUNK + lane];
      if (use_masks) {
        earlier = stM[buf * CHUNK + lane];
        have_mask = true;
      }
      mem_fence_compiler();
      if (t + NSTAGE - 1 < NCHUNKS_TOT) {  // refill pipeline tail
        const int tn = t + NSTAGE - 1;
        const int cn = tn & (NCHUNKS - 1);
        const int bn = tn & (NSTAGE - 1);
        async_cp_b32(perm + cn * CHUNK + lane, stP + bn * CHUNK + lane);
        async_cp_b32(uniforms + (size_t)tn * CHUNK + lane,
                     stU + bn * CHUNK + lane);
        if (use_masks)
          async_cp_b32(masks + cn * CHUNK + lane, stM + bn * CHUNK + lane);
      }
#else
      uv = perm[c * CHUNK + lane];
      r = uniforms[(size_t)t * CHUNK + lane];
      if (use_masks) {
        earlier = masks[c * CHUNK + lane];
        have_mask = true;
      }
      if (t + 1 < NCHUNKS_TOT) {
        const int cn = (t + 1) & (NCHUNKS - 1);
        __builtin_prefetch(perm + cn * CHUNK + lane, 0, 0);
        __builtin_prefetch(uniforms + (size_t)(t + 1) * CHUNK + lane, 0, 0);
      }
#endif
      const int u = uv >> 8;
      const int v = uv & (LYD - 1);

      if (!have_mask) {
        // Fallback: pairwise Chebyshev<=1 conflict mask vs earlier lanes.
#pragma unroll 1
        for (int j = 0; j < CHUNK; ++j) {
          int uvj = __shfl(uv, j, CHUNK);
          int uj = uvj >> 8, vj = uvj & (LYD - 1);
          int du = u - uj; du = du < 0 ? -du : du;
          int dv = v - vj; dv = dv < 0 ? -dv : dv;
          if (j < lane && du <= 1 && dv <= 1) earlier |= (1u << j);
        }
      }

      // Commit maximal conflict-free prefixes: exact sequential semantics,
      // almost always a single round for a random permutation.
      unsigned done = 0u;
      while (done != 0xFFFFFFFFu) {
        const bool active = ((done >> lane) & 1u) == 0u;
        int x = 0;
        if (active) {
          unsigned cnt = 0;  // 4 packed byte counters, one per class
#pragma unroll
          for (int dy = -1; dy <= 1; ++dy) {
#pragma unroll
            for (int dx = -1; dx <= 1; ++dx) {
              if (dx == 0 && dy == 0) continue;
              int un = u + dy; un = un < 0 ? 0 : (un > LXD - 1 ? LXD - 1 : un);
              int vn = v + dx; vn = vn < 0 ? 0 : (vn > LYD - 1 ? LYD - 1 : vn);
              unsigned nb = Xs[(un << 8) | vn];
              cnt += 1u << (nb << 3);
            }
          }
          const float w0 = __expf(beta * (float)(cnt & 0xffu));
          const float w1 = __expf(beta * (float)((cnt >> 8) & 0xffu));
          const float w2 = __expf(beta * (float)((cnt >> 16) & 0xffu));
          const float w3 = __expf(beta * (float)((cnt >> 24) & 0xffu));
          const float a1 = w0, a2 = w0 + w1, a3 = w0 + w1 + w2;
          const float thr = r * (a3 + w3);  // cumsum(softmax) < r <=> cumw < r*S
          x = (a1 < thr) + (a2 < thr) + (a3 < thr);
        }
        const bool blocked = (earlier & ~done) != 0u;
        unsigned bal = (unsigned)__ballot(active && blocked);
        unsigned q = bal ? (unsigned)(__ffs(bal) - 1) : 32u;
        unsigned pref = (q >= 32u) ? 0xFFFFFFFFu : ((1u << q) - 1u);
        unsigned committed = (~done) & pref;  // prefix of active lanes
        if (committed & (1u << lane)) {
          Xs[(u << 8) | v] = (unsigned char)x;
        }
        done |= committed;
        mem_fence_compiler();  // force reload of touched neighbours next round
      }
    }
  }
  __syncthreads();

  for (int i = tid; i < NSITES; i += BLOCKT) {
    out[i] = (float)Xs[i];
  }
}

extern "C" void kernel_launch(void* const* d_in, const int* in_sizes, int n_in,
                              void* d_out, int out_size, void* d_ws,
                              size_t ws_size, hipStream_t stream) {
  (void)in_sizes; (void)n_in; (void)out_size;
  const int* X_init = (const int*)d_in[0];
  const int* perm = (const int*)d_in[1];
  const float* uni = (const float*)d_in[2];
  const float* beta = (const float*)d_in[3];
  float* out = (float*)d_out;

  const int use_masks = (d_ws != nullptr && ws_size >= MASK_BYTES) ? 1 : 0;
  unsigned* masks = (unsigned*)d_ws;

  if (use_masks) {
    const int blocks = NCHUNKS / (BLOCKT / CHUNK);  // 2048 chunks / 8 per block
    conflict_mask_kernel<<<dim3(blocks), dim3(BLOCKT), 0, stream>>>(perm, masks);
  }

  const size_t smem =
      (size_t)NSITES + 3 * NSTAGE * CHUNK * sizeof(int);  // 67072 B
  (void)hipFuncSetAttribute((const void*)gibbs_potts_kernel,
                            hipFuncAttributeMaxDynamicSharedMemorySize,
                            (int)smem);
  gibbs_potts_kernel<<<dim3(1), dim3(BLOCKT), smem, stream>>>(
      X_init, perm, uni, beta, masks, use_masks, out);
}